// Model_76338748720023
// MI455X (gfx1250) — compile-verified
//
#include <hip/hip_runtime.h>

typedef __attribute__((ext_vector_type(2))) float v2f;
typedef __attribute__((ext_vector_type(8))) float v8f;

#define HDIM 64

// Y[N x 64] = X[N x D] @ W[D x 64]   (+ agg[row]*inv_deg[row] + bias[col]; optional relu)
// One block = 16 output rows; wave w covers cols [16w, 16w+16). D in {64,128}, N % 16 == 0.
__global__ __launch_bounds__(128) void gemm_sage_kernel(
    const float* __restrict__ X, const float* __restrict__ W,
    float* __restrict__ Y,
    const float* __restrict__ agg, const float* __restrict__ deg,
    const float* __restrict__ bias,
    int D, int relu)
{
    __shared__ float lds_w[128 * HDIM];        // up to 32 KB
    __shared__ float lds_x[16 * (128 + 2)];    // padded rows -> no bank conflicts on A

    const int tid  = threadIdx.x;
    const int wave = tid >> 5;
    const int lane = tid & 31;
    const int hh   = lane >> 4;     // half-wave select
    const int l    = lane & 15;
    const int row0 = blockIdx.x * 16;
    const int col0 = wave * 16;
    const int Dp   = D + 2;
    const int dsh  = (D == 128) ? 7 : 6;

    // cooperative stage: W (D*64 contiguous) and X tile (16 contiguous rows)
    for (int i = tid; i < D * HDIM; i += 128) lds_w[i] = W[i];
    const float* Xt = X + (size_t)row0 * D;
    for (int i = tid; i < 16 * D; i += 128) {
        int r = i >> dsh;
        int c = i & (D - 1);
        lds_x[r * Dp + c] = Xt[i];
    }
    __syncthreads();

    v8f acc = {0.f, 0.f, 0.f, 0.f, 0.f, 0.f, 0.f, 0.f};
    const int koff = 2 * hh;        // lanes 0-15: K=k,k+1 ; lanes 16-31: K=k+2,k+3
    const int c    = col0 + l;

    for (int k = 0; k < D; k += 4) {
        v2f a = *(const v2f*)&lds_x[l * Dp + k + koff];   // A: X[row0+l][k+koff .. +1]
        v2f b;
        b.x = lds_w[(k + koff) * HDIM + c];               // B: W[k+koff][c]
        b.y = lds_w[(k + koff + 1) * HDIM + c];
        acc = __builtin_amdgcn_wmma_f32_16x16x4_f32(
                  false, a, false, b, (short)0, acc, false, false);
    }

#pragma unroll
    for (int m = 0; m < 8; ++m) {
        int row = row0 + hh * 8 + m;                      // C/D layout: lane half picks M+8
        float v = acc[m];
        if (agg) {
            float inv = 1.0f / fmaxf(deg[row], 1.0f);
            v += agg[(size_t)row * HDIM + c] * inv;
        }
        if (bias) v += bias[c];
        if (relu) v = fmaxf(v, 0.0f);
        Y[(size_t)row * HDIM + c] = v;
    }
}

__global__ void degree_kernel(const int* __restrict__ src, const int* __restrict__ dst,
                              float* __restrict__ deg_u, float* __restrict__ deg_p, int E)
{
    int e = blockIdx.x * blockDim.x + threadIdx.x;
    if (e < E) {
        atomicAdd(&deg_u[src[e]], 1.0f);
        atomicAdd(&deg_p[dst[e]], 1.0f);
    }
}

// acc[sidx[e]] += feat[gidx[e]]  (64 floats/edge; 16 threads/edge, float4 each)
__global__ void scatter_kernel(const float* __restrict__ feat,
                               const int* __restrict__ gidx, const int* __restrict__ sidx,
                               float* __restrict__ acc, int E)
{
    int tid = blockIdx.x * blockDim.x + threadIdx.x;
    int e = tid >> 4;
    if (e < E) {
        int j = (tid & 15) * 4;
        int g = gidx[e], s = sidx[e];
        float4 v = *(const float4*)(feat + (size_t)g * HDIM + j);
        float* a = acc + (size_t)s * HDIM + j;
        atomicAdd(a + 0, v.x);
        atomicAdd(a + 1, v.y);
        atomicAdd(a + 2, v.z);
        atomicAdd(a + 3, v.w);
    }
}

__global__ void dot_kernel(const float* __restrict__ hu, const float* __restrict__ hp,
                           const int* __restrict__ lu, const int* __restrict__ lp,
                           float* __restrict__ out, int n)
{
    int i = blockIdx.x * blockDim.x + threadIdx.x;
    if (i < n) {
        const float4* a = (const float4*)(hu + (size_t)lu[i] * HDIM);
        const float4* b = (const float4*)(hp + (size_t)lp[i] * HDIM);
        float s = 0.0f;
#pragma unroll
        for (int j = 0; j < 16; ++j) {
            float4 x = a[j], y = b[j];
            s += x.x * y.x + x.y * y.y + x.z * y.z + x.w * y.w;
        }
        out[i] = s;
    }
}

extern "C" void kernel_launch(void* const* d_in, const int* in_sizes, int n_in,
                              void* d_out, int out_size, void* d_ws, size_t ws_size,
                              hipStream_t stream) {
    const float* x_user    = (const float*)d_in[0];
    const float* x_product = (const float*)d_in[1];
    const float* W1_buys_l = (const float*)d_in[2];
    const float* b1_buys   = (const float*)d_in[3];
    const float* W1_buys_r = (const float*)d_in[4];
    const float* W1_rev_l  = (const float*)d_in[5];
    const float* b1_rev    = (const float*)d_in[6];
    const float* W1_rev_r  = (const float*)d_in[7];
    const float* W2_buys_l = (const float*)d_in[8];
    const float* b2_buys   = (const float*)d_in[9];
    const float* W2_buys_r = (const float*)d_in[10];
    const float* W2_rev_l  = (const float*)d_in[11];
    const float* b2_rev    = (const float*)d_in[12];
    const float* W2_rev_r  = (const float*)d_in[13];
    const int*   edge_src  = (const int*)d_in[14];
    const int*   edge_dst  = (const int*)d_in[15];
    const int*   lab_u     = (const int*)d_in[16];
    const int*   lab_p     = (const int*)d_in[17];

    const int N_U = in_sizes[0] / 64;    // 100000
    const int N_P = in_sizes[1] / 128;   // 50000
    const int E     = in_sizes[14];      // 2000000
    const int E_LAB = in_sizes[16];      // 500000

    float* ws = (float*)d_ws;
    size_t off = 0;
    float* deg_u = ws + off; off += (size_t)N_U;
    float* deg_p = ws + off; off += (size_t)N_P;
    float* agg_u = ws + off; off += (size_t)N_U * HDIM;
    float* agg_p = ws + off; off += (size_t)N_P * HDIM;
    float* t_u   = ws + off; off += (size_t)N_U * HDIM;   // yu1 / yu2, reused as h_u2
    float* t_p   = ws + off; off += (size_t)N_P * HDIM;   // yp1 / yp2, reused as h_p2
    float* h_u   = ws + off; off += (size_t)N_U * HDIM;
    float* h_p   = ws + off; off += (size_t)N_P * HDIM;

    // zero degrees + accumulators (contiguous region)
    hipMemsetAsync(deg_u, 0,
                   ((size_t)(N_U + N_P) + (size_t)(N_U + N_P) * HDIM) * sizeof(float),
                   stream);

    const dim3 gblk(128);
    const int gu = N_U / 16, gp = N_P / 16;
    const int egrid = (E + 255) / 256;
    const int sgrid = (E * 16 + 255) / 256;

    // ---- layer 1: transform-before-aggregate (seg_mean(x)@W == seg_mean(x@W)) ----
    gemm_sage_kernel<<<gu, gblk, 0, stream>>>(x_user,    W1_buys_l, t_u, nullptr, nullptr, nullptr, 64, 0);
    gemm_sage_kernel<<<gp, gblk, 0, stream>>>(x_product, W1_rev_l,  t_p, nullptr, nullptr, nullptr, 128, 0);

    degree_kernel<<<egrid, 256, 0, stream>>>(edge_src, edge_dst, deg_u, deg_p, E);

    scatter_kernel<<<sgrid, 256, 0, stream>>>(t_u, edge_src, edge_dst, agg_p, E);
    scatter_kernel<<<sgrid, 256, 0, stream>>>(t_p, edge_dst, edge_src, agg_u, E);

    // h = relu(agg/deg + bias + x_dst @ W_r)   (fused epilogue)
    gemm_sage_kernel<<<gp, gblk, 0, stream>>>(x_product, W1_buys_r, h_p, agg_p, deg_p, b1_buys, 128, 1);
    gemm_sage_kernel<<<gu, gblk, 0, stream>>>(x_user,    W1_rev_r,  h_u, agg_u, deg_u, b1_rev,  64,  1);

    // ---- layer 2 ----
    gemm_sage_kernel<<<gu, gblk, 0, stream>>>(h_u, W2_buys_l, t_u, nullptr, nullptr, nullptr, 64, 0);
    gemm_sage_kernel<<<gp, gblk, 0, stream>>>(h_p, W2_rev_l,  t_p, nullptr, nullptr, nullptr, 64, 0);

    hipMemsetAsync(agg_u, 0, (size_t)(N_U + N_P) * HDIM * sizeof(float), stream);

    scatter_kernel<<<sgrid, 256, 0, stream>>>(t_u, edge_src, edge_dst, agg_p, E);
    scatter_kernel<<<sgrid, 256, 0, stream>>>(t_p, edge_dst, edge_src, agg_u, E);

    // h_*2 written into t_* (their previous contents were fully consumed above)
    gemm_sage_kernel<<<gp, gblk, 0, stream>>>(h_p, W2_buys_r, t_p, agg_p, deg_p, b2_buys, 64, 0);
    gemm_sage_kernel<<<gu, gblk, 0, stream>>>(h_u, W2_rev_r,  t_u, agg_u, deg_u, b2_rev,  64, 0);

    // ---- classifier: per-label-edge dot ----
    dot_kernel<<<(E_LAB + 255) / 256, 256, 0, stream>>>(t_u, t_p, lab_u, lab_p, (float*)d_out, E_LAB);
}